// ModifiedRAFT_43198781063936
// MI455X (gfx1250) — compile-verified
//
#include <hip/hip_runtime.h>

typedef float v2f __attribute__((ext_vector_type(2)));
typedef float v8f __attribute__((ext_vector_type(8)));

#define FD 128          // feature dim
#define NK 81           // (2*4+1)^2 offsets
#define WIN 10          // per-pixel integer window extent

// ---------------------------------------------------------------------------
// avg_pool2 for BOTH feature maps in one launch. Channels = FD.
// ---------------------------------------------------------------------------
__global__ void pool2_pair_kernel(const float* __restrict__ in1, const float* __restrict__ in2,
                                  float* __restrict__ out1, float* __restrict__ out2,
                                  int Hi, int Wi) {
    int Ho = Hi >> 1, Wo = Wi >> 1;
    int total = FD * Ho * Wo;
    int idx = blockIdx.x * blockDim.x + threadIdx.x;
    if (idx >= 2 * total) return;
    const float* in = (idx < total) ? in1 : in2;
    float* out      = (idx < total) ? out1 : out2;
    int i = (idx < total) ? idx : idx - total;
    int w = i % Wo;
    int h = (i / Wo) % Ho;
    int d = i / (Wo * Ho);
    const float* q = in + ((size_t)d * Hi + 2 * h) * Wi + 2 * w;
    out[i] = 0.25f * (q[0] + q[1] + q[Wi] + q[Wi + 1]);
}

// ---------------------------------------------------------------------------
// jax.image.resize(coords,(B,2,Ho,Wo),'linear'), antialias triangle kernel
// stretched by f=2^lvl, per-dim renormalization at borders. All 3 levels fused.
// Input (2,64,96).
// ---------------------------------------------------------------------------
__global__ void resize_coords_all_kernel(const float* __restrict__ in,
                                         float* __restrict__ o1, float* __restrict__ o2,
                                         float* __restrict__ o3) {
    const int Hi = 64, Wi = 96;
    int idx = blockIdx.x * blockDim.x + threadIdx.x;
    if (idx >= 3072 + 768 + 192) return;
    int lvl, local; float* out;
    if (idx < 3072)      { lvl = 1; local = idx;        out = o1; }
    else if (idx < 3840) { lvl = 2; local = idx - 3072; out = o2; }
    else                 { lvl = 3; local = idx - 3840; out = o3; }
    int Ho = Hi >> lvl, Wo = Wi >> lvl;
    float f = (float)(1 << lvl);
    int w = local % Wo;
    int h = (local / Wo) % Ho;
    int ch = local / (Wo * Ho);
    float cyc = (h + 0.5f) * f - 0.5f;
    float cxc = (w + 0.5f) * f - 0.5f;
    int yA = max(0, (int)ceilf(cyc - f)), yB = min(Hi - 1, (int)floorf(cyc + f));
    int xA = max(0, (int)ceilf(cxc - f)), xB = min(Wi - 1, (int)floorf(cxc + f));
    const float* src = in + (size_t)ch * Hi * Wi;
    float sumy = 0.0f, sumx = 0.0f, acc = 0.0f;
    for (int xx = xA; xx <= xB; ++xx) {
        float wx = 1.0f - fabsf((float)xx - cxc) / f;
        if (wx > 0.0f) sumx += wx;
    }
    for (int yy = yA; yy <= yB; ++yy) {
        float wy = 1.0f - fabsf((float)yy - cyc) / f;
        if (wy <= 0.0f) continue;
        sumy += wy;
        float rowacc = 0.0f;
        for (int xx = xA; xx <= xB; ++xx) {
            float wx = 1.0f - fabsf((float)xx - cxc) / f;
            if (wx > 0.0f) rowacc += wx * src[(size_t)yy * Wi + xx];
        }
        acc += wy * rowacc;
    }
    out[local] = acc / (sumy * sumx);
}

// ---------------------------------------------------------------------------
// Per-tile correlation, templated on level dims so all feature-map strides
// are immediate offsets (no 64-bit address chains in the hot loop).
// ---------------------------------------------------------------------------
template <int H, int W>
__device__ __forceinline__ void corr_tile(
    const float* __restrict__ f1, const float* __restrict__ f2,
    const float* __restrict__ coords, float* __restrict__ out,
    float cscale, int tile, int lane,
    float* s_cx, float* s_cy, int* s_bx, int* s_by, float* s_dot) {
    constexpr int HW = H * W;
    const int j = lane & 15;                 // pixel slot / B column
    const bool hi = lane >= 16;
    const int kd = hi ? 2 : 0;               // K sub-index per ISA A/B layout

    const int p  = tile * 16 + j;            // HW is a multiple of 16 at all levels
    const int ph = p / W, pw = p % W;

    if (!hi) {
        float cx = coords[ph * W + pw] * cscale;
        float cy = coords[HW + ph * W + pw] * cscale;
        float lx = fminf(fmaxf(cx - 4.0f, 0.0f), (float)(W - 1));
        float ly = fminf(fmaxf(cy - 4.0f, 0.0f), (float)(H - 1));
        s_cx[j] = cx; s_cy[j] = cy;
        s_bx[j] = (int)floorf(lx);
        s_by[j] = (int)floorf(ly);
    }
    __syncthreads();

    // Union bounding box of the 16 per-pixel windows (uniform across the wave).
    int Xmin = W, Xmax = 0, Ymin = H, Ymax = 0;
    for (int t = 0; t < 16; ++t) {
        int bx = s_bx[t], by = s_by[t];
        Xmin = min(Xmin, bx); Xmax = max(Xmax, bx + WIN - 1);
        Ymin = min(Ymin, by); Ymax = max(Ymax, by + WIN - 1);
    }
    Xmax = min(Xmax, W - 1);
    Ymax = min(Ymax, H - 1);

    // Per-lane window bases for the C rows this lane owns (M = r + 8*hi).
    int bxm[8], bym[8];
#pragma unroll
    for (int r = 0; r < 8; ++r) {
        int m = r + (hi ? 8 : 0);
        bxm[r] = s_bx[m]; bym[r] = s_by[m];
    }

    // Stage A = f1 tile in WMMA 16x4-f32 layout (constant-immediate offsets):
    //   VGPR0: K=0 (lanes 0-15) / K=2 (lanes 16-31); VGPR1: K=1 / K=3.
    v2f a[32];
    {
        const float* abase = f1 + (kd * H + ph) * W + pw;
#pragma unroll
        for (int kk = 0; kk < 32; ++kk) {
            a[kk].x = abase[(4 * kk) * HW];
            a[kk].y = abase[(4 * kk + 1) * HW];
        }
    }

    for (int y = Ymin; y <= Ymax; ++y) {
        if (y + 1 <= Ymax)  // warm next f2 row (global_prefetch_b8)
            __builtin_prefetch(f2 + (kd * H + (y + 1)) * W + Xmin, 0, 0);
        for (int Xc = Xmin; Xc <= Xmax; Xc += 16) {
            int x = Xc + j; if (x > W - 1) x = W - 1;  // clamp; dup scatter idempotent
            const float* bbase = f2 + (kd * H + y) * W + x;
            // Batch all B loads first (independent, immediate offsets -> clauses).
            v2f b[32];
#pragma unroll
            for (int kk = 0; kk < 32; ++kk) {
                b[kk].x = bbase[(4 * kk) * HW];
                b[kk].y = bbase[(4 * kk + 1) * HW];
            }
            // Two independent accumulation chains for MMA ILP.
            v8f c0 = {0.f, 0.f, 0.f, 0.f, 0.f, 0.f, 0.f, 0.f};
            v8f c1 = {0.f, 0.f, 0.f, 0.f, 0.f, 0.f, 0.f, 0.f};
#pragma unroll
            for (int kk = 0; kk < 32; kk += 2) {
                c0 = __builtin_amdgcn_wmma_f32_16x16x4_f32(
                         false, a[kk],     false, b[kk],     (short)0, c0, false, false);
                c1 = __builtin_amdgcn_wmma_f32_16x16x4_f32(
                         false, a[kk + 1], false, b[kk + 1], (short)0, c1, false, false);
            }
            // Scatter C into per-pixel 10x10 dot windows.
#pragma unroll
            for (int r = 0; r < 8; ++r) {
                int relx = x - bxm[r];
                int rely = y - bym[r];
                if ((unsigned)relx < (unsigned)WIN && (unsigned)rely < (unsigned)WIN) {
                    int m = r + (hi ? 8 : 0);
                    s_dot[m * (WIN * WIN) + rely * WIN + relx] = c0[r] + c1[r];
                }
            }
        }
    }
    __syncthreads();

    // Combine: 81 offsets x 16 pixels, bilinear weights from the dot tables.
    const float invSqrtD = 0.08838834764831845f;  // 1/sqrt(128)
    for (int e = lane; e < 16 * NK; e += 32) {
        int jj = e / NK;
        int k  = e % NK;
        int ix = k / 9, iy = k % 9;                // dx slow, dy fast (meshgrid 'ij')
        float ox = (float)(ix - 4), oy = (float)(iy - 4);
        float cx = s_cx[jj], cy = s_cy[jj];
        float xx = fminf(fmaxf(cx + ox, 0.0f), (float)(W - 1));
        float yy = fminf(fmaxf(cy + oy, 0.0f), (float)(H - 1));
        float x0f = floorf(xx), y0f = floorf(yy);
        int x0 = (int)x0f, y0 = (int)y0f;
        int x1 = min(x0 + 1, W - 1), y1 = min(y0 + 1, H - 1);
        float wx1 = xx - x0f, wx0 = 1.0f - wx1;
        float wy1 = yy - y0f, wy0 = 1.0f - wy1;
        int bx = s_bx[jj], by = s_by[jj];
        const float* dw = &s_dot[jj * (WIN * WIN)];
        float d00 = dw[(y0 - by) * WIN + (x0 - bx)];
        float d10 = dw[(y0 - by) * WIN + (x1 - bx)];
        float d01 = dw[(y1 - by) * WIN + (x0 - bx)];
        float d11 = dw[(y1 - by) * WIN + (x1 - bx)];
        float corr = (wx0 * wy0 * d00 + wx1 * wy0 * d10 +
                      wx0 * wy1 * d01 + wx1 * wy1 * d11) * invSqrtD;
        int pp = tile * 16 + jj;
        out[k * HW + pp] = corr;
    }
}

// ---------------------------------------------------------------------------
// All 4 levels in one launch: blocks [0,384) L0, [384,480) L1, [480,504) L2,
// [504,510) L3. One wave32 per block; LDS shared across instantiations.
// ---------------------------------------------------------------------------
__global__ __launch_bounds__(32) void corr_all_kernel(
    const float* f1l0, const float* f2l0, const float* c0p, float* o0,
    const float* f1l1, const float* f2l1, const float* c1p, float* o1,
    const float* f1l2, const float* f2l2, const float* c2p, float* o2,
    const float* f1l3, const float* f2l3, const float* c3p, float* o3) {
    __shared__ float s_cx[16], s_cy[16];
    __shared__ int   s_bx[16], s_by[16];
    __shared__ float s_dot[16 * WIN * WIN];
    const int lane = threadIdx.x;
    const int b = blockIdx.x;
    if (b < 384) {
        corr_tile<64, 96>(f1l0, f2l0, c0p, o0, 1.0f,   b,       lane, s_cx, s_cy, s_bx, s_by, s_dot);
    } else if (b < 480) {
        corr_tile<32, 48>(f1l1, f2l1, c1p, o1, 0.5f,   b - 384, lane, s_cx, s_cy, s_bx, s_by, s_dot);
    } else if (b < 504) {
        corr_tile<16, 24>(f1l2, f2l2, c2p, o2, 0.25f,  b - 480, lane, s_cx, s_cy, s_bx, s_by, s_dot);
    } else {
        corr_tile< 8, 12>(f1l3, f2l3, c3p, o3, 0.125f, b - 504, lane, s_cx, s_cy, s_bx, s_by, s_dot);
    }
}

// ---------------------------------------------------------------------------
extern "C" void kernel_launch(void* const* d_in, const int* in_sizes, int n_in,
                              void* d_out, int out_size, void* d_ws, size_t ws_size,
                              hipStream_t stream) {
    (void)in_sizes; (void)n_in; (void)out_size; (void)ws_size;
    const float* fmap1  = (const float*)d_in[0];   // (1,128,64,96)
    const float* fmap2  = (const float*)d_in[1];
    const float* coords = (const float*)d_in[2];   // (1,2,64,96)
    float* out = (float*)d_out;
    float* ws  = (float*)d_ws;

    // Workspace layout (floats): pooled pyramids + resized coords (~2.1 MB total)
    float* f1l1 = ws;                 // 128*32*48 = 196608
    float* f1l2 = f1l1 + 196608;      // 128*16*24 =  49152
    float* f1l3 = f1l2 + 49152;       // 128* 8*12 =  12288
    float* f2l1 = f1l3 + 12288;
    float* f2l2 = f2l1 + 196608;
    float* f2l3 = f2l2 + 49152;
    float* cl1  = f2l3 + 12288;       // 2*32*48 = 3072
    float* cl2  = cl1 + 3072;         // 2*16*24 =  768
    float* cl3  = cl2 + 768;          // 2* 8*12 =  192

    auto pool = [&](const float* i1, const float* i2, float* q1, float* q2, int Hi, int Wi) {
        int total = 2 * FD * (Hi / 2) * (Wi / 2);
        pool2_pair_kernel<<<(total + 255) / 256, 256, 0, stream>>>(i1, i2, q1, q2, Hi, Wi);
    };
    pool(fmap1, fmap2, f1l1, f2l1, 64, 96);
    pool(f1l1, f2l1, f1l2, f2l2, 32, 48);
    pool(f1l2, f2l2, f1l3, f2l3, 16, 24);

    resize_coords_all_kernel<<<(4032 + 255) / 256, 256, 0, stream>>>(coords, cl1, cl2, cl3);

    // Outputs concatenated flat in return order: (1,81,Hl,Wl) per level.
    float* out0 = out;
    float* out1 = out0 + (size_t)NK * 64 * 96;
    float* out2 = out1 + (size_t)NK * 32 * 48;
    float* out3 = out2 + (size_t)NK * 16 * 24;

    corr_all_kernel<<<510, 32, 0, stream>>>(
        fmap1, fmap2, coords, out0,
        f1l1,  f2l1,  cl1,    out1,
        f1l2,  f2l2,  cl2,    out2,
        f1l3,  f2l3,  cl3,    out3);
}